// NashLoss2D_7301444403971
// MI455X (gfx1250) — compile-verified
//
#include <hip/hip_runtime.h>
#include <cstdint>

// Problem constants (from reference)
#define NT 10000
#define NS 5000

// Main-kernel geometry: 5 blocks * 8 waves * 32 lanes * 4 cols = 5120 cols (>= 5000)
#define NXBLK 5
#define WAVES 8
#define DEPTH 8                      // async pipeline depth (rows in flight per wave)
#define NCOL4PAD (NXBLK * WAVES * 32) // 1280 column-quads
#define STAGE2_BLOCKS ((NS + 255) / 256) // 20

typedef __attribute__((ext_vector_type(2))) float v2f;
typedef __attribute__((ext_vector_type(8))) float v8f;

// ---- CDNA5 async global->LDS path (ASYNCcnt-tracked DMA into LDS) -----------
#define S_WAIT_ASYNCCNT(n) asm volatile("s_wait_asynccnt %0" ::"n"(n) : "memory")

__device__ __forceinline__ void async_b128(uint32_t lds_byte_off, uint32_t g_byte_off,
                                           const float* sbase) {
  // GVS mode: mem = SGPR64 base + VGPR32 offset ; VDST = LDS byte address VGPR
  asm volatile("global_load_async_to_lds_b128 %0, %1, %2"
               :: "v"(lds_byte_off), "v"(g_byte_off),
                  "s"((unsigned long long)(uintptr_t)sbase)
               : "memory");
}

// ---- Wave-wide f32 sum via V_WMMA_F32_16X16X4_F32 ---------------------------
// A = ones(16x4)  =>  D[m][n] = sum_k B[k][n]  (identical for every row m).
// Summing c[0] over all 32 lanes counts every B element exactly twice
// (lanes 0-15 hold row M=0, lanes 16-31 hold row M=8 of the same D), so
// total = shfl-sum * 0.5. Exact in f32, independent of B's lane->(k,n) map.
__device__ __forceinline__ float wave_sum32(float x) {
#if __has_builtin(__builtin_amdgcn_wmma_f32_16x16x4_f32)
  v2f a; a.x = 1.0f; a.y = 1.0f;          // all-ones A (16x4)
  v2f b; b.x = x;    b.y = 0.0f;          // 32 lane values + 32 zeros
  v8f c = {0.f, 0.f, 0.f, 0.f, 0.f, 0.f, 0.f, 0.f};
  c = __builtin_amdgcn_wmma_f32_16x16x4_f32(false, a, false, b, (short)0, c,
                                            false, false);
  float v = c[0];
#pragma unroll
  for (int off = 16; off > 0; off >>= 1) v += __shfl_xor(v, off, 32);
  return v * 0.5f;
#else
#pragma unroll
  for (int off = 16; off > 0; off >>= 1) x += __shfl_xor(x, off, 32);
  return x;
#endif
}

// ---- Stage 1: streaming masked reduction, async 8-deep ring buffer through LDS -
__global__ __launch_bounds__(256) void nash_stage1(
    const float* __restrict__ pred, const float* __restrict__ targ,
    float4* __restrict__ ws4, int rpc) {
  __shared__ __align__(16) float ldsbuf[WAVES * DEPTH * 256]; // 64 KB / block

  const int tid  = threadIdx.x;
  const int wave = tid >> 5;
  const int lane = tid & 31;
  const int c4   = blockIdx.x * 256 + wave * 32 + lane;  // 0..1279 (unique)
  int col = c4 * 4;
  if (col > NS - 4) col = NS - 4;  // clamp tail quads in-bounds; results unused

  const int chunk = blockIdx.y;
  const int r0 = chunk * rpc;
  int r1 = r0 + rpc; if (r1 > NT) r1 = NT;
  const int rows = (r1 > r0) ? (r1 - r0) : 0;

  float* __restrict__ slotbase = &ldsbuf[wave * DEPTH * 256];
  const uint32_t ldsLane = (uint32_t)(uintptr_t)slotbase + (uint32_t)lane * 16u;

  float acc_cnt[4] = {0.f, 0.f, 0.f, 0.f};
  float acc_st [4] = {0.f, 0.f, 0.f, 0.f};
  float acc_s2 [4] = {0.f, 0.f, 0.f, 0.f};
  float acc_rs [4] = {0.f, 0.f, 0.f, 0.f};

  const uint32_t rowStride = (uint32_t)NS * 4u;
  uint32_t goff = ((uint32_t)r0 * (uint32_t)NS + (uint32_t)col) * 4u;

  // Prologue: fill the pipeline (2 async ops per row: pred + targ)
  const int pre = rows < DEPTH ? rows : DEPTH;
  for (int i = 0; i < pre; ++i) {
    const uint32_t so = ldsLane + (uint32_t)i * 1024u;
    async_b128(so,        goff, pred);
    async_b128(so + 512u, goff, targ);
    goff += rowStride;
  }

  int steady = rows - DEPTH; if (steady < 0) steady = 0;
  int slot = 0;

#define NASH_ACC(j, pv_, tv_)                                                 \
  {                                                                           \
    const float tt = (tv_); const float pp = (pv_);                           \
    const bool  m  = (tt == tt); /* !isnan */                                 \
    const float tm = m ? tt : 0.f;                                            \
    const float pm = m ? pp : 0.f;                                            \
    acc_cnt[j] += m ? 1.f : 0.f;                                              \
    acc_st[j]  += tm;                                                         \
    acc_s2[j]   = fmaf(tm, tm, acc_s2[j]);                                    \
    const float d = tm - pm;                                                  \
    acc_rs[j]   = fmaf(d, d, acc_rs[j]);                                      \
  }

  for (int i = 0; i < steady; ++i) {
    // Oldest row's 2 ops complete once ASYNCcnt <= 2*(DEPTH-1) (in-order done)
    S_WAIT_ASYNCCNT(2 * (DEPTH - 1));
    const float4 pv = *(const float4*)&slotbase[slot * 256 + lane * 4];
    const float4 tv = *(const float4*)&slotbase[slot * 256 + 128 + lane * 4];
    NASH_ACC(0, pv.x, tv.x); NASH_ACC(1, pv.y, tv.y);
    NASH_ACC(2, pv.z, tv.z); NASH_ACC(3, pv.w, tv.w);
    const uint32_t so = ldsLane + (uint32_t)slot * 1024u;  // reuse freed slot
    async_b128(so,        goff, pred);
    async_b128(so + 512u, goff, targ);
    goff += rowStride;
    slot = (slot + 1 == DEPTH) ? 0 : slot + 1;
  }
  S_WAIT_ASYNCCNT(0);  // drain tail
  for (int i = steady; i < rows; ++i) {
    const float4 pv = *(const float4*)&slotbase[slot * 256 + lane * 4];
    const float4 tv = *(const float4*)&slotbase[slot * 256 + 128 + lane * 4];
    NASH_ACC(0, pv.x, tv.x); NASH_ACC(1, pv.y, tv.y);
    NASH_ACC(2, pv.z, tv.z); NASH_ACC(3, pv.w, tv.w);
    slot = (slot + 1 == DEPTH) ? 0 : slot + 1;
  }
#undef NASH_ACC

  // Deterministic per-chunk partials: 4 stat-quads per column-quad (64 B)
  float4* o = ws4 + ((size_t)chunk * NCOL4PAD + (size_t)c4) * 4;
  o[0] = make_float4(acc_cnt[0], acc_cnt[1], acc_cnt[2], acc_cnt[3]);
  o[1] = make_float4(acc_st [0], acc_st [1], acc_st [2], acc_st [3]);
  o[2] = make_float4(acc_s2 [0], acc_s2 [1], acc_s2 [2], acc_s2 [3]);
  o[3] = make_float4(acc_rs [0], acc_rs [1], acc_rs [2], acc_rs [3]);
}

// ---- Stage 2: fold chunks per column, per-column NSE term, block reduce -----
__global__ __launch_bounds__(256) void nash_stage2(
    const float* __restrict__ wsP, float2* __restrict__ bsum, int nch) {
  const int tid = threadIdx.x;
  const int col = blockIdx.x * 256 + tid;
  float pc = 0.f, nv = 0.f;
  if (col < NS) {
    const int c4 = col >> 2, j = col & 3;
    float cnt = 0.f, st = 0.f, s2 = 0.f, rs = 0.f;
    for (int ch = 0; ch < nch; ++ch) {
      const float* e = wsP + ((size_t)ch * NCOL4PAD + (size_t)c4) * 16 + j;
      cnt += e[0]; st += e[4]; s2 += e[8]; rs += e[12];
    }
    const float n1   = cnt > 1.f ? cnt : 1.f;
    const float mean = st / n1;
    const float sst  = s2 - cnt * mean * mean; // == sum_valid (t-mean)^2
    const bool  valid = (cnt > 10.f) && (sst != 0.f);
    const float den  = sqrtf(valid ? sst : 1.f) + 0.1f;
    pc = valid ? rs / (den * den) : 0.f;
    nv = valid ? 1.f : 0.f;
  }
  // EXEC fully reconverged here (required for WMMA)
  const float wpc = wave_sum32(pc);
  const float wnv = wave_sum32(nv);
  __shared__ float spc[WAVES], snv[WAVES];
  const int wave = tid >> 5, lane = tid & 31;
  if (lane == 0) { spc[wave] = wpc; snv[wave] = wnv; }
  __syncthreads();
  if (tid == 0) {
    float a = 0.f, b = 0.f;
    for (int w = 0; w < WAVES; ++w) { a += spc[w]; b += snv[w]; }
    bsum[blockIdx.x] = make_float2(a, b);
  }
}

// ---- Stage 3: single wave, WMMA-reduce block partials, emit scalar ----------
__global__ void nash_stage3(const float2* __restrict__ bsum, int nblk,
                            float* __restrict__ out) {
  const int lane = threadIdx.x & 31;
  float pc = 0.f, nv = 0.f;
  if (lane < nblk) { const float2 v = bsum[lane]; pc = v.x; nv = v.y; }
  const float tp = wave_sum32(pc);
  const float tn = wave_sum32(nv);
  if (lane == 0) out[0] = tp / tn;
}

// ---- Host launcher ----------------------------------------------------------
extern "C" void kernel_launch(void* const* d_in, const int* in_sizes, int n_in,
                              void* d_out, int out_size, void* d_ws, size_t ws_size,
                              hipStream_t stream) {
  (void)in_sizes; (void)n_in; (void)out_size;
  const float* pred = (const float*)d_in[0];
  const float* targ = (const float*)d_in[1];
  float* out = (float*)d_out;

  // Per-chunk partial footprint: 1280 quads * 16 floats * 4 B = 81920 B
  const size_t perChunk = (size_t)NCOL4PAD * 16 * sizeof(float);
  int nch = 128;  // 640 blocks -> plenty of waves to fill HBM pipeline
  while (nch > 1 &&
         ((size_t)nch * perChunk + STAGE2_BLOCKS * sizeof(float2)) > ws_size)
    nch >>= 1;
  const int rpc = (NT + nch - 1) / nch;

  float4* ws4  = (float4*)d_ws;
  float2* bsum = (float2*)((char*)d_ws + (size_t)nch * perChunk);

  dim3 g1(NXBLK, nch);
  nash_stage1<<<g1, dim3(256), 0, stream>>>(pred, targ, ws4, rpc);
  nash_stage2<<<STAGE2_BLOCKS, 256, 0, stream>>>((const float*)d_ws, bsum, nch);
  nash_stage3<<<1, 32, 0, stream>>>(bsum, STAGE2_BLOCKS, out);
}